// Coo2Cel_231928234119
// MI455X (gfx1250) — compile-verified
//
#include <hip/hip_runtime.h>

// MI455X / gfx1250 implementation.
//
// Problem shape (fixed by reference setup): B=1, N=1024, S=27 shifts.
// Output = [N,N,S,4] f32 = 453 MB  ->  pure streaming-store bound
// (~19 us floor at 23.3 TB/s). Strategy:
//   Kernel 1 (64 blocks x 1 wave): pos = pos_cel @ cel via
//     V_WMMA_F32_16X16X4_F32 (the reference einsum, K padded 3->4),
//     result stored as float4 (w=0); also builds the 27 shift vectors with
//     the pbc "allowed" flag packed in .w.
//   Kernel 2 (grid 108 x 1024): i = blockIdx.y (scalar-loaded position),
//     inner = j*27+s from blockIdx.x*256+t (only a cheap 16-bit div-by-27),
//     tables staged into LDS via CDNA5 GLOBAL_LOAD_ASYNC_TO_LDS_B128,
//     output written with coalesced non-temporal global_store_b128 so the
//     453 MB write-once stream doesn't thrash the 192 MB L2.

typedef float v2f __attribute__((ext_vector_type(2)));
typedef float v4f __attribute__((ext_vector_type(4)));
typedef float v8f __attribute__((ext_vector_type(8)));

#define N_ATOMS 1024
#define N_SHIFT 27
#define INNER   (N_ATOMS * N_SHIFT)           // 27648 = 108 * 256
#define RC2     36.0f                         // rc = 6.0

// ---------------------------------------------------------------------------
// Kernel 1: Cartesian positions via WMMA f32 16x16x4 (einsum 'nc,cd->nd').
// One wave32 per 16-row tile. A = pos_cel rows (16x4, K=3 padded with 0),
// B = cel matrix (4x16, only cols 0..2 nonzero). D column n (n<4) lives in
// lane n / lane 16+n; rows r and r+8 in C/D VGPRs 0..7.
// ---------------------------------------------------------------------------
__global__ __launch_bounds__(32) void pos_einsum_wmma(
    const float* __restrict__ pos_cel,     // [N,3] fractional
    const float* __restrict__ cel,         // [3,3] row-major
    const unsigned char* __restrict__ pbc, // [3] bool bytes
    v4f* __restrict__ pos_ws,              // [N] float4 (x,y,z,0)
    v4f* __restrict__ sft_ws)              // [27] float4 (sx,sy,sz,allowed)
{
  const int lane = threadIdx.x;   // 0..31
  const int tile = blockIdx.x;    // 0..63
  const int m    = lane & 15;     // A row within tile / B-D column index
  const int half = lane >> 4;     // K-pair select per ISA 32-bit A layout
  const int k0   = half * 2;      // {0,2}
  const int k1   = k0 + 1;        // {1,3}
  const int row  = tile * 16 + m;

  v2f a, b;
  a.x = pos_cel[row * 3 + k0];                       // k0 in {0,2} valid
  a.y = (k1 < 3) ? pos_cel[row * 3 + k1] : 0.0f;     // k=3 pad
  b.x = (m < 3) ? cel[k0 * 3 + m] : 0.0f;
  b.y = (k1 < 3 && m < 3) ? cel[k1 * 3 + m] : 0.0f;

  v8f c = {0.f, 0.f, 0.f, 0.f, 0.f, 0.f, 0.f, 0.f};
  // (neg_a, A, neg_b, B, c_mod, C, reuse_a, reuse_b)
  c = __builtin_amdgcn_wmma_f32_16x16x4_f32(false, a, false, b, (short)0, c,
                                            false, false);

  // Scatter D: lane holds column m for rows (half*8 + r), r=0..7.
  if (m < 4) {
#pragma unroll
    for (int r = 0; r < 8; ++r) {
      const int M = tile * 16 + half * 8 + r;
      ((float*)(pos_ws + M))[m] = c[r];
    }
  }

  // Block 0 also builds the 27 shift vectors: sft_xyz[s] = s_int @ cel,
  // with the pbc-allowed predicate folded into .w.
  if (tile == 0 && lane < N_SHIFT) {
    const int sx = lane / 9 - 1;
    const int sy = (lane / 3) % 3 - 1;
    const int sz = lane % 3 - 1;
    const float fx = (float)sx, fy = (float)sy, fz = (float)sz;
    v4f s4;
    s4.x = fx * cel[0] + fy * cel[3] + fz * cel[6];
    s4.y = fx * cel[1] + fy * cel[4] + fz * cel[7];
    s4.z = fx * cel[2] + fy * cel[5] + fz * cel[8];
    const bool allowed = (sx == 0 || pbc[0] != 0) &&
                         (sy == 0 || pbc[1] != 0) &&
                         (sz == 0 || pbc[2] != 0);
    s4.w = allowed ? 1.0f : 0.0f;
    sft_ws[lane] = s4;
  }
}

// ---------------------------------------------------------------------------
// Kernel 2: grid (INNER/256, N). One thread per (j,s) for row i=blockIdx.y.
// Consecutive lanes -> consecutive float4 stores (coalesced NT b128).
// ---------------------------------------------------------------------------
__global__ __launch_bounds__(256) void pairs_kernel(
    const v4f* __restrict__ pos_ws,
    const v4f* __restrict__ sft_ws,
    v4f* __restrict__ out)
{
  __shared__ v4f lds_pos[N_ATOMS];  // 16 KB
  __shared__ v4f lds_sft[32];

  const unsigned t = threadIdx.x;

  // --- Stage tables into LDS via CDNA5 async copies (ASYNCcnt-tracked,
  //     bypasses VGPRs). Low 32 bits of a generic LDS pointer are the
  //     wave-relative LDS byte address (flat-LDS aperture truncation).
  {
    const unsigned lds_p = (unsigned)(uintptr_t)(&lds_pos[0]);
    const char*    gp    = (const char*)pos_ws;
#pragma unroll
    for (int u = 0; u < 4; ++u) {
      const unsigned idx = t + 256u * (unsigned)u;
      const unsigned           laddr = lds_p + idx * 16u;
      const unsigned long long gaddr =
          (unsigned long long)(uintptr_t)(gp + idx * 16u);
      asm volatile("global_load_async_to_lds_b128 %0, %1, off"
                   :: "v"(laddr), "v"(gaddr) : "memory");
    }
    if (t < N_SHIFT) {
      const unsigned           laddr =
          (unsigned)(uintptr_t)(&lds_sft[0]) + t * 16u;
      const unsigned long long gaddr =
          (unsigned long long)(uintptr_t)(sft_ws + t);
      asm volatile("global_load_async_to_lds_b128 %0, %1, off"
                   :: "v"(laddr), "v"(gaddr) : "memory");
    }
    asm volatile("s_wait_asynccnt 0x0" ::: "memory");
  }
  __syncthreads();

  const unsigned i     = blockIdx.y;             // pair row (block-uniform)
  const unsigned inner = blockIdx.x * 256u + t;  // 0..27647 = j*27 + s
  const unsigned j     = inner / (unsigned)N_SHIFT;  // 16-bit magic divide
  const unsigned s     = inner - j * (unsigned)N_SHIFT;

  const v4f pi = pos_ws[i];    // uniform address -> scalar load (s_load_b128)
  const v4f pj = lds_pos[j];   // ~uniform within wave (LDS broadcast)
  const v4f sf = lds_sft[s];

  const float vx = pi.x - pj.x - sf.x;
  const float vy = pi.y - pj.y - sf.y;
  const float vz = pi.z - pj.z - sf.z;
  const float sod = vx * vx + vy * vy + vz * vz;

  // mask = in-cutoff & shift-allowed & not-self-pair-at-zero-shift (s==13)
  const bool self = (i == j) & (s == 13u);
  const bool mask = (sod < RC2) && (sf.w != 0.0f) && !self;

  v4f o;
  o.x = mask ? vx : 0.0f;
  o.y = mask ? vy : 0.0f;
  o.z = mask ? vz : 0.0f;
  o.w = mask ? sod : 0.0f;

  // Write-once stream: non-temporal so 453 MB doesn't evict useful L2 lines.
  __builtin_nontemporal_store(o, out + (size_t)i * INNER + inner);
}

// ---------------------------------------------------------------------------
extern "C" void kernel_launch(void* const* d_in, const int* in_sizes, int n_in,
                              void* d_out, int out_size, void* d_ws,
                              size_t ws_size, hipStream_t stream)
{
  (void)in_sizes; (void)n_in; (void)out_size; (void)ws_size;
  const float* pos_cel = (const float*)d_in[0];             // [1,1024,3]
  const float* cel     = (const float*)d_in[1];             // [1,3,3]
  const unsigned char* pbc = (const unsigned char*)d_in[2]; // [1,3] bool

  v4f* pos_ws = (v4f*)d_ws;                                  // 16 KB
  v4f* sft_ws = (v4f*)((char*)d_ws + N_ATOMS * sizeof(v4f)); // 432 B
  v4f* out    = (v4f*)d_out;

  hipLaunchKernelGGL(pos_einsum_wmma, dim3(N_ATOMS / 16), dim3(32), 0, stream,
                     pos_cel, cel, pbc, pos_ws, sft_ws);

  hipLaunchKernelGGL(pairs_kernel, dim3(INNER / 256, N_ATOMS), dim3(256), 0,
                     stream, pos_ws, sft_ws, out);
}